// SVM_MVO_Net_50414326120539
// MI455X (gfx1250) — compile-verified
//
#include <hip/hip_runtime.h>
#include <hip/hip_bf16.h>
#include <math.h>

// ---------------------------------------------------------------------------
// SVM + MVO differentiable pipeline, MI455X (gfx1250, wave32, WMMA).
//
// Factorization: K = (F W^T)(F W^T)^T = F (W^T W) F^T = F G F^T, G is dxd
// (256x256).  K@x is computed as F(G(F^T x)) -> no 64MB K matrix, SVM PGD
// becomes L2-resident skinny matvecs.  eta_svm needs row sums of |K|, done by
// a streaming fp32 WMMA GEMM (H F^T, H = F G) that reduces |tile| rows with
// lane shuffles + atomicAdd and never stores K.  MVO matvecs stream Sigma
// (64MB, resident in the 192MB L2).  All GEMM math uses V_WMMA_F32_16X16X4_F32
// to keep fp32 accuracy for the QP.  Row-major fragment loads are issued as
// single 8-byte (b64) loads to halve VMEM issue count in the WMMA loops.
// ---------------------------------------------------------------------------

typedef float v2f __attribute__((ext_vector_type(2)));
typedef float v8f __attribute__((ext_vector_type(8)));

#define C_SVM 1.0f
#define TAU_S 0.01f
#define PGD_ITERS 150
#define BISECT_ITERS 60
#define NEWTON_ITERS 2
#define NB_FTX 16

// ---------------- WMMA fragment helpers (fp32 16x16x4) ----------------------
// A (16x4, MxK) per ISA: lane l<16 -> (M=l, K=k0+0/1 in v0/v1),
//                        lane l>=16 -> (M=l-16, K=k0+2/3).
// Row-major source base[row][k]; contiguous pair -> one global_load_b64.
__device__ __forceinline__ v2f load_fragA(const float* __restrict__ base,
                                          int row0, int k0, int ld) {
  int lane = threadIdx.x & 31;
  const float* p = base + (size_t)(row0 + (lane & 15)) * ld + k0 + ((lane >> 4) << 1);
  return *(const v2f*)p;  // 8B-aligned: k0 % 4 == 0, half-offset is 0 or 2
}
// B (4x16, KxN): lane l<16 -> (N=n0+l, K=k0+0/1), lane l>=16 -> (N=n0+l-16, K=k0+2/3).
// Row-major KxN source base[k][n]: elements come from two rows (strided).
__device__ __forceinline__ v2f load_fragB(const float* __restrict__ base,
                                          int n0, int k0, int ld) {
  int lane = threadIdx.x & 31;
  int col = n0 + (lane & 15);
  int kk = k0 + ((lane >> 4) << 1);
  v2f f; f.x = base[(size_t)kk * ld + col]; f.y = base[(size_t)(kk + 1) * ld + col];
  return f;
}
// C/D 16x16 f32: VGPR r -> lanes0-15: (M=r, N=lane), lanes16-31: (M=r+8, N=lane-16).
__device__ __forceinline__ void store_tile(float* __restrict__ base,
                                           int m0, int n0, int ld, v8f c) {
  int lane = threadIdx.x & 31;
  int col = n0 + (lane & 15);
  int mo = m0 + ((lane >> 4) << 3);
#pragma unroll
  for (int r = 0; r < 8; ++r) base[(size_t)(mo + r) * ld + col] = c[r];
}

__device__ __forceinline__ v8f wmma_f32(v2f a, v2f b, v8f c) {
  return __builtin_amdgcn_wmma_f32_16x16x4_f32(false, a, false, b, (short)0, c,
                                               false, false);
}

// ---------------- reductions (wave32) ---------------------------------------
__device__ __forceinline__ float warpSum(float v) {
#pragma unroll
  for (int o = 16; o; o >>= 1) v += __shfl_xor(v, o, 32);
  return v;
}
__device__ __forceinline__ float warpMax(float v) {
#pragma unroll
  for (int o = 16; o; o >>= 1) v = fmaxf(v, __shfl_xor(v, o, 32));
  return v;
}
__device__ __forceinline__ float warpMin(float v) {
#pragma unroll
  for (int o = 16; o; o >>= 1) v = fminf(v, __shfl_xor(v, o, 32));
  return v;
}
__device__ float blockSum(float v, float* sh) {
  __syncthreads();
  v = warpSum(v);
  int wid = threadIdx.x >> 5, lane = threadIdx.x & 31, nw = blockDim.x >> 5;
  if (lane == 0) sh[wid] = v;
  __syncthreads();
  if (wid == 0) {
    float r = (lane < nw) ? sh[lane] : 0.0f;
    r = warpSum(r);
    if (lane == 0) sh[0] = r;
  }
  __syncthreads();
  return sh[0];
}
__device__ float blockMax(float v, float* sh) {
  __syncthreads();
  v = warpMax(v);
  int wid = threadIdx.x >> 5, lane = threadIdx.x & 31, nw = blockDim.x >> 5;
  if (lane == 0) sh[wid] = v;
  __syncthreads();
  if (wid == 0) {
    float r = (lane < nw) ? sh[lane] : -3.0e38f;
    r = warpMax(r);
    if (lane == 0) sh[0] = r;
  }
  __syncthreads();
  return sh[0];
}
__device__ float blockMin(float v, float* sh) {
  __syncthreads();
  v = warpMin(v);
  int wid = threadIdx.x >> 5, lane = threadIdx.x & 31, nw = blockDim.x >> 5;
  if (lane == 0) sh[wid] = v;
  __syncthreads();
  if (wid == 0) {
    float r = (lane < nw) ? sh[lane] : 3.0e38f;
    r = warpMin(r);
    if (lane == 0) sh[0] = r;
  }
  __syncthreads();
  return sh[0];
}

__device__ __forceinline__ float ylab(int i) { return (i & 1) ? 1.0f : -1.0f; }
__device__ __forceinline__ float clipf(float x, float lo, float hi) {
  return fminf(fmaxf(x, lo), hi);
}

// ---------------- WMMA GEMM kernels -----------------------------------------
// G[d x d] = W^T W, inner dim n.  A = W^T (fragB from W), B = W (fragB).
__global__ void k_gemm_WtW(const float* __restrict__ W, float* __restrict__ G,
                           int n, int d) {
  int wid = threadIdx.x >> 5;
  int tile = blockIdx.x * (blockDim.x >> 5) + wid;
  int tn = d >> 4;
  int m0 = (tile / tn) << 4, n0 = (tile % tn) << 4;
  v8f c = {};
  for (int k0 = 0; k0 < n; k0 += 4) {
    v2f a = load_fragB(W, m0, k0, d);  // A[m][k] = W[k][m]
    v2f b = load_fragB(W, n0, k0, d);  // B[k][n] = W[k][n]
    c = wmma_f32(a, b, c);
  }
  store_tile(G, m0, n0, d, c);
}

// H[n x d] = F G, inner dim d.
__global__ void k_gemm_FG(const float* __restrict__ F, const float* __restrict__ G,
                          float* __restrict__ H, int n, int d) {
  int wid = threadIdx.x >> 5;
  int tile = blockIdx.x * (blockDim.x >> 5) + wid;
  int tn = d >> 4;
  int m0 = (tile / tn) << 4, n0 = (tile % tn) << 4;
  v8f c = {};
  for (int k0 = 0; k0 < d; k0 += 4) {
    v2f a = load_fragA(F, m0, k0, d);
    v2f b = load_fragB(G, n0, k0, d);
    c = wmma_f32(a, b, c);
  }
  store_tile(H, m0, n0, d, c);
}

// Streaming row sums of |K| where K = H F^T (inner dim d); K tiles never stored.
__global__ void k_rowabsK(const float* __restrict__ H, const float* __restrict__ F,
                          float* __restrict__ rowsum, int n, int d) {
  int wid = threadIdx.x >> 5;
  int tile = blockIdx.x * (blockDim.x >> 5) + wid;
  int tn = n >> 4;
  int m0 = (tile / tn) << 4, n0 = (tile % tn) << 4;
  v8f c = {};
  for (int k0 = 0; k0 < d; k0 += 4) {
    v2f a = load_fragA(H, m0, k0, d);
    v2f b = load_fragA(F, n0, k0, d);  // B = F^T -> fragA pattern on F rows
    c = wmma_f32(a, b, c);
  }
  int lane = threadIdx.x & 31;
  int mo = m0 + ((lane >> 4) << 3);
#pragma unroll
  for (int r = 0; r < 8; ++r) {
    float s = fabsf(c[r]);
    s += __shfl_xor(s, 1, 16);
    s += __shfl_xor(s, 2, 16);
    s += __shfl_xor(s, 4, 16);
    s += __shfl_xor(s, 8, 16);
    if ((lane & 15) == 0) atomicAdd(&rowsum[mo + r], s);
  }
}

// ---------------- vector / matvec kernels -----------------------------------
__global__ void k_zero(float* __restrict__ p, int n) {
  int i = blockIdx.x * blockDim.x + threadIdx.x;
  if (i < n) p[i] = 0.0f;
}

// out[row] = sum_j M[row][j] * x[j]   (one wave per row, float4 lanes)
__global__ void k_matvec(const float* __restrict__ M, const float* __restrict__ x,
                         float* __restrict__ out, int n) {
  int wid = threadIdx.x >> 5, lane = threadIdx.x & 31;
  int row = blockIdx.x * (blockDim.x >> 5) + wid;
  const float* mp = M + (size_t)row * n;
  float acc = 0.0f;
  for (int j = lane * 4; j < n; j += 128) {
    float4 mv = *(const float4*)(mp + j);
    float4 xv = *(const float4*)(x + j);
    acc += mv.x * xv.x + mv.y * xv.y + mv.z * xv.z + mv.w * xv.w;
  }
  acc = warpSum(acc);
  if (lane == 0) out[row] = acc;
}

// out[row] = sum_j |M[row][j]|
__global__ void k_rowabs(const float* __restrict__ M, float* __restrict__ out, int n) {
  int wid = threadIdx.x >> 5, lane = threadIdx.x & 31;
  int row = blockIdx.x * (blockDim.x >> 5) + wid;
  const float* mp = M + (size_t)row * n;
  float acc = 0.0f;
  for (int j = lane * 4; j < n; j += 128) {
    float4 mv = *(const float4*)(mp + j);
    acc += fabsf(mv.x) + fabsf(mv.y) + fabsf(mv.z) + fabsf(mv.w);
  }
  acc = warpSum(acc);
  if (lane == 0) out[row] = acc;
}

// scal[idx] = 1 / (max_i rowsum[i] + 1)
__global__ void k_eta(const float* __restrict__ rowsum, float* __restrict__ scal,
                      int n, int idx) {
  __shared__ float sh[32];
  float v = 0.0f;
  for (int i = threadIdx.x; i < n; i += blockDim.x) v = fmaxf(v, rowsum[i]);
  v = blockMax(v, sh);
  if (threadIdx.x == 0) scal[idx] = 1.0f / (v + 1.0f);
}

// part[b][j] = sum_{i in block-range} (a[i]*y[i]) * F[i][j]   (F^T (y.*a) partials)
__global__ void k_ftx(const float* __restrict__ F, const float* __restrict__ a,
                      float* __restrict__ part, int n, int d) {
  __shared__ float ay[256];
  int j = threadIdx.x;
  int rows = n / gridDim.x;
  int r0 = blockIdx.x * rows;
  float acc = 0.0f;
  for (int c0 = 0; c0 < rows; c0 += blockDim.x) {
    int i = r0 + c0 + threadIdx.x;
    ay[threadIdx.x] = a[i] * ylab(i);
    __syncthreads();
    for (int q = 0; q < blockDim.x; ++q)
      acc += ay[q] * F[(size_t)(r0 + c0 + q) * d + j];
    __syncthreads();
  }
  part[blockIdx.x * d + j] = acc;
}

// u = sum_b part[b]; vv = G u        (single block of d threads)
__global__ void k_gu(const float* __restrict__ part, const float* __restrict__ G,
                     float* __restrict__ vv, int d, int nb) {
  __shared__ float ush[256];
  int j = threadIdx.x;
  float u = 0.0f;
  for (int b = 0; b < nb; ++b) u += part[b * d + j];
  ush[j] = u;
  __syncthreads();
  const float* g = G + (size_t)j * d;
  float acc = 0.0f;
  for (int k = 0; k < d; ++k) acc += g[k] * ush[k];
  vv[j] = acc;
}

// t[row] = F[row][:] . vv   (one wave per row; vv cached in LDS)
__global__ void k_fv(const float* __restrict__ F, const float* __restrict__ vv,
                     float* __restrict__ t, int n, int d) {
  __shared__ float vsh[256];
  if (threadIdx.x < d) vsh[threadIdx.x] = vv[threadIdx.x];
  __syncthreads();
  int wid = threadIdx.x >> 5, lane = threadIdx.x & 31;
  int row = blockIdx.x * (blockDim.x >> 5) + wid;
  const float* fp = F + (size_t)row * d;
  float acc = 0.0f;
  for (int j = lane * 4; j < d; j += 128) {
    float4 fvv = *(const float4*)(fp + j);
    acc += fvv.x * vsh[j] + fvv.y * vsh[j + 1] + fvv.z * vsh[j + 2] + fvv.w * vsh[j + 3];
  }
  acc = warpSum(acc);
  if (lane == 0) t[row] = acc;
}

// ---------------- projections (single workgroup) ----------------------------
// a <- proj_{0<=a<=C, y.a=0}( a - eta*(y.*t - 1) ),  t = K(y.*a)
__global__ void k_svm_proj(const float* __restrict__ t, float* __restrict__ a,
                           const float* __restrict__ scal, int n) {
  __shared__ float sh[32];
  float eta = scal[0];
  int items = n / blockDim.x;  // 4 for n=4096, 1024 threads
  float v[8], yv[8];
  float mx = 0.0f;
  for (int q = 0; q < items; ++q) {
    int i = threadIdx.x + q * blockDim.x;
    yv[q] = ylab(i);
    v[q] = a[i] - eta * (yv[q] * t[i] - 1.0f);
    mx = fmaxf(mx, fabsf(v[q]));
  }
  float bound = blockMax(mx, sh) + C_SVM;
  float lo = -bound, hi = bound;
  for (int it = 0; it < BISECT_ITERS; ++it) {
    float mid = 0.5f * (lo + hi);
    float g = 0.0f;
    for (int q = 0; q < items; ++q)
      g += yv[q] * clipf(v[q] - mid * yv[q], 0.0f, C_SVM);
    g = blockSum(g, sh);
    if (g > 0.0f) lo = mid; else hi = mid;
  }
  float lam = 0.5f * (lo + hi);
  for (int nt = 0; nt < NEWTON_ITERS; ++nt) {
    float s = 0.0f, cnt = 0.0f;
    for (int q = 0; q < items; ++q) {
      float u = v[q] - lam * yv[q];
      float aa = clipf(u, 0.0f, C_SVM);
      s += yv[q] * aa;
      cnt += (u > 0.0f && u < C_SVM) ? 1.0f : 0.0f;
    }
    s = blockSum(s, sh);
    cnt = blockSum(cnt, sh);
    lam += s / fmaxf(cnt, 1.0f);
  }
  for (int q = 0; q < items; ++q) {
    int i = threadIdx.x + q * blockDim.x;
    a[i] = clipf(v[q] - lam * yv[q], 0.0f, C_SVM);
  }
}

// w <- proj_{0<=w<=m, sum w=1}( init? 1/n : w - eta*(t - lam*mu) ); lam update.
__global__ void k_mvo_proj(const float* __restrict__ t, const float* __restrict__ mu,
                           const float* __restrict__ mcap, float* __restrict__ w,
                           float* __restrict__ scal, int n, int init) {
  __shared__ float sh[32];
  float eta = scal[1];
  float lam = init ? 0.0f : scal[2];
  int items = n / blockDim.x;
  float v[8], mc[8];
  float vmin = 3.0e38f, vmax = -3.0e38f, mmax = -3.0e38f;
  for (int q = 0; q < items; ++q) {
    int i = threadIdx.x + q * blockDim.x;
    mc[q] = mcap[i];
    v[q] = init ? (1.0f / (float)n) : (w[i] - eta * (t[i] - lam * mu[i]));
    vmin = fminf(vmin, v[q]);
    vmax = fmaxf(vmax, v[q]);
    mmax = fmaxf(mmax, mc[q]);
  }
  vmin = blockMin(vmin, sh);
  vmax = blockMax(vmax, sh);
  mmax = blockMax(mmax, sh);
  float lo = vmin - mmax - 1.0f, hi = vmax;
  for (int it = 0; it < BISECT_ITERS; ++it) {
    float mid = 0.5f * (lo + hi);
    float r = 0.0f;
    for (int q = 0; q < items; ++q) r += clipf(v[q] - mid, 0.0f, mc[q]);
    r = blockSum(r, sh) - 1.0f;
    if (r > 0.0f) lo = mid; else hi = mid;
  }
  float tau = 0.5f * (lo + hi);
  for (int nt = 0; nt < NEWTON_ITERS; ++nt) {
    float s = 0.0f, cnt = 0.0f;
    for (int q = 0; q < items; ++q) {
      float u = v[q] - tau;
      s += clipf(u, 0.0f, mc[q]);
      cnt += (u > 0.0f && u < mc[q]) ? 1.0f : 0.0f;
    }
    s = blockSum(s, sh);
    cnt = blockSum(cnt, sh);
    tau += (s - 1.0f) / fmaxf(cnt, 1.0f);
  }
  float rdot = 0.0f;
  for (int q = 0; q < items; ++q) {
    int i = threadIdx.x + q * blockDim.x;
    float ww = clipf(v[q] - tau, 0.0f, mc[q]);
    w[i] = ww;
    rdot += mu[i] * ww;
  }
  rdot = blockSum(rdot, sh);
  if (threadIdx.x == 0)
    scal[2] = init ? 0.0f : fmaxf(lam + eta * (0.0f - rdot), 0.0f);
}

// sv = first free support vector; b = y[sv]-margins[sv]; m = sigmoid((margins+b)/tau)
__global__ void k_finalize(const float* __restrict__ margins, const float* __restrict__ a,
                           float* __restrict__ mcap, int n) {
  __shared__ int shi[32];
  int best = 0x7fffffff;
  for (int i = threadIdx.x; i < n; i += blockDim.x) {
    float av = a[i];
    if (av > 1e-4f && av < C_SVM - 1e-4f && i < best) best = i;
  }
#pragma unroll
  for (int o = 16; o; o >>= 1) best = min(best, __shfl_xor(best, o, 32));
  int wid = threadIdx.x >> 5, lane = threadIdx.x & 31, nw = blockDim.x >> 5;
  if (lane == 0) shi[wid] = best;
  __syncthreads();
  if (wid == 0) {
    int b2 = (lane < nw) ? shi[lane] : 0x7fffffff;
#pragma unroll
    for (int o = 16; o; o >>= 1) b2 = min(b2, __shfl_xor(b2, o, 32));
    if (lane == 0) shi[0] = b2;
  }
  __syncthreads();
  int sv = shi[0];
  if (sv == 0x7fffffff) sv = 0;
  float b = ylab(sv) - margins[sv];
  for (int i = threadIdx.x; i < n; i += blockDim.x)
    mcap[i] = 1.0f / (1.0f + expf(-(margins[i] + b) / TAU_S));
}

__global__ void k_loss(const float* __restrict__ w, const float* __restrict__ t,
                       float* __restrict__ out, int n) {
  __shared__ float sh[32];
  float s = 0.0f;
  for (int i = threadIdx.x; i < n; i += blockDim.x) s += w[i] * t[i];
  s = blockSum(s, sh);
  if (threadIdx.x == 0) out[0] = 0.5f * s;
}

__global__ void k_copy_out(const float* __restrict__ a, const float* __restrict__ m,
                           const float* __restrict__ w, float* __restrict__ out, int n) {
  int i = blockIdx.x * blockDim.x + threadIdx.x;
  if (i < n) out[1 + i] = a[i];
  else if (i < 2 * n) out[1 + i] = m[i - n];
  else if (i < 3 * n) out[1 + i] = w[i - 2 * n];
}

// ---------------- launcher ---------------------------------------------------
extern "C" void kernel_launch(void* const* d_in, const int* in_sizes, int n_in,
                              void* d_out, int out_size, void* d_ws, size_t ws_size,
                              hipStream_t stream) {
  const float* F = (const float*)d_in[0];      // feat_mat (n x d)
  const float* Sigma = (const float*)d_in[1];  // (n x n)
  const float* mu = (const float*)d_in[2];     // (n)
  const float* W = (const float*)d_in[3];      // (n x d)
  int n = in_sizes[2];
  int d = in_sizes[0] / n;

  float* ws = (float*)d_ws;
  float* G = ws;                                // d*d
  float* H = G + (size_t)d * d;                 // n*d
  float* part = H + (size_t)n * d;              // NB_FTX*d
  float* vv = part + NB_FTX * d;                // d
  float* t = vv + d;                            // n
  float* a = t + n;                             // n
  float* w = a + n;                             // n
  float* mcap = w + n;                          // n
  float* rowsum = mcap + n;                     // n
  float* scal = rowsum + n;                     // scal[0]=eta_svm [1]=eta_mvo [2]=lam

  int td = d >> 4, tn = n >> 4;

  // G = W^T W ; H = F G     (fp32 WMMA)
  k_gemm_WtW<<<(td * td) / 8, 256, 0, stream>>>(W, G, n, d);
  k_gemm_FG<<<(tn * td) / 8, 256, 0, stream>>>(F, G, H, n, d);

  // eta_svm from streaming |K| row sums (K = H F^T, never stored)
  k_zero<<<(n + 255) / 256, 256, 0, stream>>>(rowsum, n);
  k_rowabsK<<<(tn * tn) / 8, 256, 0, stream>>>(H, F, rowsum, n, d);
  k_eta<<<1, 1024, 0, stream>>>(rowsum, scal, n, 0);

  // eta_mvo from Sigma row sums
  k_rowabs<<<n / 8, 256, 0, stream>>>(Sigma, rowsum, n);
  k_eta<<<1, 1024, 0, stream>>>(rowsum, scal, n, 1);

  // SVM dual PGD: a=0; repeat t = F(G(F^T(y.*a))), project.
  k_zero<<<(n + 255) / 256, 256, 0, stream>>>(a, n);
  for (int it = 0; it < PGD_ITERS; ++it) {
    k_ftx<<<NB_FTX, d, 0, stream>>>(F, a, part, n, d);
    k_gu<<<1, d, 0, stream>>>(part, G, vv, d, NB_FTX);
    k_fv<<<n / 8, 256, 0, stream>>>(F, vv, t, n, d);
    k_svm_proj<<<1, 1024, 0, stream>>>(t, a, scal, n);
  }

  // margins = K(y.*a); b from first free SV; m = sigmoid((margins+b)/tau)
  k_ftx<<<NB_FTX, d, 0, stream>>>(F, a, part, n, d);
  k_gu<<<1, d, 0, stream>>>(part, G, vv, d, NB_FTX);
  k_fv<<<n / 8, 256, 0, stream>>>(F, vv, t, n, d);
  k_finalize<<<1, 1024, 0, stream>>>(t, a, mcap, n);

  // MVO primal-dual PGD (Sigma stays L2-resident across 150 matvecs)
  k_mvo_proj<<<1, 1024, 0, stream>>>(t, mu, mcap, w, scal, n, 1);
  for (int it = 0; it < PGD_ITERS; ++it) {
    k_matvec<<<n / 8, 256, 0, stream>>>(Sigma, w, t, n);
    k_mvo_proj<<<1, 1024, 0, stream>>>(t, mu, mcap, w, scal, n, 0);
  }

  // loss = 0.5 w' Sigma w ; pack outputs [loss, a, m, w]
  k_matvec<<<n / 8, 256, 0, stream>>>(Sigma, w, t, n);
  k_loss<<<1, 1024, 0, stream>>>(w, t, (float*)d_out, n);
  k_copy_out<<<(3 * n + 255) / 256, 256, 0, stream>>>(a, mcap, w, (float*)d_out, n);
}